// DAGGenome_32908039422013
// MI455X (gfx1250) — compile-verified
//
#include <hip/hip_runtime.h>

// DAG reachability + subtree-leaf-flag kernel for MI455X (gfx1250, wave32).
//
//  * Forward-pointing edges (child > parent): reachability = one ascending
//    sweep, subtree flags = one descending sweep. Wave 0 and wave 1 run the
//    two sweeps concurrently on different SIMDs (disjoint LDS state;
//    same-wave DS ops are in-order, so no barriers inside the sweeps).
//  * State is bit-packed: active = 1 u32 per 32-node tile, flags = 1 u64 per
//    tile (score ballot | reroll ballot << 32). Seeds are one uniform LDS
//    word load, results one ballot store, scatters are ds_or_b32 atomics.
//  * In-tile edges are rare (~6% of tiles have any): the shuffle-closure and
//    OR-reduces are gated behind __any() so the common path is ~1 DS hop.
//  * left/right are staged through LDS with global_load_async_to_lds_b32
//    (ASYNCcnt), 16-slot ring, issue depth 8 tiles, s_wait_asynccnt 14
//    (async loads complete in order -> oldest tile guaranteed resident).

#define N_NODES 16384
#define NGROUPS (N_NODES / 32)
#define RING    16
#define DEPTH   8

__device__ __forceinline__ unsigned lds_off(const void* p) {
    // Generic LDS pointer: low 32 bits are the workgroup-relative LDS offset.
    return (unsigned)(unsigned long long)(uintptr_t)p;
}

__device__ __forceinline__ void async_b32(const int* gaddr, unsigned loff) {
    asm volatile("global_load_async_to_lds_b32 %0, %1, off"
                 :: "v"(loff), "v"((unsigned long long)(uintptr_t)gaddr)
                 : "memory");
}

__device__ __forceinline__ void wait_async_le14() {
    asm volatile("s_wait_asynccnt 14" ::: "memory");
}
__device__ __forceinline__ void wait_async_0() {
    asm volatile("s_wait_asynccnt 0" ::: "memory");
}

__global__ __launch_bounds__(256)
void dag_genome_kernel(const int* __restrict__ left,
                       const int* __restrict__ right,
                       const unsigned char* __restrict__ leaf_is_reroll,
                       float* __restrict__ out)
{
    __shared__ unsigned           active_words[NGROUPS];   // 2 KB, bit per node
    __shared__ unsigned long long flags64[NGROUPS];        // 4 KB, score|reroll<<32
    __shared__ int stageFL[RING][32], stageFR[RING][32];   // 4 KB fwd staging
    __shared__ int stageBL[RING][32], stageBR[RING][32];   // 4 KB bwd staging
    __shared__ int lds_count;

    const int tid  = threadIdx.x;
    const int wave = tid >> 5;
    const int lane = tid & 31;

    for (int i = tid; i < NGROUPS; i += 256) active_words[i] = 0u;
    __syncthreads();

    if (wave == 0) {
        // ---------------- Forward pass: reachability from node 0 ----------------
        #pragma unroll
        for (int t = 0; t < DEPTH; ++t) {
            async_b32(&left [t * 32 + lane], lds_off(&stageFL[t][lane]));
            async_b32(&right[t * 32 + lane], lds_off(&stageFR[t][lane]));
        }
        int cnt = 0;
        for (int g = 0; g < NGROUPS; ++g) {
            wait_async_le14();                       // tile g's operands resident
            const int slot  = g & (RING - 1);
            const int l     = stageFL[slot][lane];
            const int r     = stageFR[slot][lane];
            const int gbase = g << 5;

            unsigned seed = active_words[g];         // uniform ds_load_b32
            if (g == 0) seed |= 1u;                  // root node 0

            unsigned edges = 0u;                     // in-tile edges (higher lanes)
            if (l >= gbase && l < gbase + 32) edges |= 1u << (l - gbase);
            if (r >= gbase && r < gbase + 32) edges |= 1u << (r - gbase);

            unsigned reach_word;
            if (__any(edges != 0u)) {                // rare (~6% of tiles)
                unsigned m = edges;                  // wave-uniform target union
                #pragma unroll
                for (int off = 16; off >= 1; off >>= 1)
                    m |= (unsigned)__shfl_xor((int)m, off, 32);
                // Transitive closure, descending target order: lane b's closure
                // is final before any lane reads it (edges only go up).
                unsigned closure = edges;
                while (m) {
                    const int b = 31 - __clz((int)m);
                    m &= ~(1u << b);
                    const unsigned cb = (unsigned)__shfl((int)closure, b, 32);
                    if ((closure >> b) & 1u) closure |= cb;
                }
                unsigned contrib = ((seed >> lane) & 1u) ? closure : 0u;
                #pragma unroll
                for (int off = 16; off >= 1; off >>= 1)
                    contrib |= (unsigned)__shfl_xor((int)contrib, off, 32);
                reach_word = seed | contrib;
            } else {
                reach_word = seed;                   // no in-tile propagation
            }

            if (lane == 0) active_words[g] = reach_word;
            cnt += __popc(reach_word);               // uniform accumulate

            if ((reach_word >> lane) & 1u) {         // scatter to later tiles
                if (l >= gbase + 32) atomicOr(&active_words[l >> 5], 1u << (l & 31));
                if (r >= gbase + 32) atomicOr(&active_words[r >> 5], 1u << (r & 31));
            }

            const int nt = g + DEPTH;                // refill the ring
            if (nt < NGROUPS) {
                const int ns = nt & (RING - 1);
                async_b32(&left [nt * 32 + lane], lds_off(&stageFL[ns][lane]));
                async_b32(&right[nt * 32 + lane], lds_off(&stageFR[ns][lane]));
            }
        }
        wait_async_0();
        if (lane == 0) lds_count = cnt;
    } else if (wave == 1) {
        // ---------------- Backward pass: subtree leaf flags ----------------
        // Warm the 16 KB leaf array (random gather target) into the caches.
        for (int i = lane; i < N_NODES / 64; i += 32)
            __builtin_prefetch(&leaf_is_reroll[i << 6], 0, 3);
        #pragma unroll
        for (int t = 0; t < DEPTH; ++t) {
            const int gg = NGROUPS - 1 - t;
            const int ss = gg & (RING - 1);
            async_b32(&left [gg * 32 + lane], lds_off(&stageBL[ss][lane]));
            async_b32(&right[gg * 32 + lane], lds_off(&stageBR[ss][lane]));
        }
        for (int g = NGROUPS - 1; g >= 0; --g) {
            wait_async_le14();
            const int slot  = g & (RING - 1);
            const int l     = stageBL[slot][lane];
            const int r     = stageBR[slot][lane];
            const int gbase = g << 5;

            int s = 0, rr = 0;      // has_score / has_reroll
            int ll = -1, rl = -1;   // in-tile child lanes

            if (l < 0) {
                const int isr = leaf_is_reroll[-l - 1] ? 1 : 0;
                s |= isr ^ 1; rr |= isr;
            } else if (l >= gbase + 32) {            // later tile: final in LDS
                const unsigned long long cw = flags64[l >> 5];
                s  |= (int)((cw >> (l & 31)) & 1ull);
                rr |= (int)((cw >> (32 + (l & 31))) & 1ull);
            } else {
                ll = l - gbase;                      // in-tile, ll > lane
            }
            if (r < 0) {
                const int isr = leaf_is_reroll[-r - 1] ? 1 : 0;
                s |= isr ^ 1; rr |= isr;
            } else if (r >= gbase + 32) {
                const unsigned long long cw = flags64[r >> 5];
                s  |= (int)((cw >> (r & 31)) & 1ull);
                rr |= (int)((cw >> (32 + (r & 31))) & 1ull);
            } else {
                rl = r - gbase;
            }

            if (__any((ll >= 0) | (rl >= 0))) {      // rare in-tile children
                unsigned m = 0u;
                if (ll >= 0) m |= 1u << ll;
                if (rl >= 0) m |= 1u << rl;
                #pragma unroll
                for (int off = 16; off >= 1; off >>= 1)
                    m |= (unsigned)__shfl_xor((int)m, off, 32);
                // Descending lane order: lane b final before being read.
                while (m) {
                    const int b = 31 - __clz((int)m);
                    m &= ~(1u << b);
                    const int sb = __shfl(s,  b, 32);
                    const int rb = __shfl(rr, b, 32);
                    if (ll == b) { s |= sb; rr |= rb; }
                    if (rl == b) { s |= sb; rr |= rb; }
                }
            }

            const unsigned sw = (unsigned)__ballot(s);
            const unsigned rw = (unsigned)__ballot(rr);
            if (lane == 0)
                flags64[g] = ((unsigned long long)rw << 32) | (unsigned long long)sw;

            const int nt = g - DEPTH;
            if (nt >= 0) {
                const int ns = nt & (RING - 1);
                async_b32(&left [nt * 32 + lane], lds_off(&stageBL[ns][lane]));
                async_b32(&right[nt * 32 + lane], lds_off(&stageBR[ns][lane]));
            }
        }
        wait_async_0();
    }
    __syncthreads();

    // ---------------- Emit: [active | has_score | has_reroll | count] ----------
    const int count = lds_count;
    #pragma unroll 4
    for (int i = tid; i < N_NODES; i += 256) {
        const unsigned           w  = active_words[i >> 5];
        const unsigned long long fw = flags64[i >> 5];
        const int b = i & 31;
        out[i]               = (float)((w  >> b) & 1u);
        out[N_NODES + i]     = (float)((fw >> b) & 1ull);
        out[2 * N_NODES + i] = (float)((fw >> (32 + b)) & 1ull);
    }
    if (tid == 0) out[3 * N_NODES] = (float)count;
}

extern "C" void kernel_launch(void* const* d_in, const int* in_sizes, int n_in,
                              void* d_out, int out_size, void* d_ws, size_t ws_size,
                              hipStream_t stream) {
    (void)in_sizes; (void)n_in; (void)out_size; (void)d_ws; (void)ws_size;
    // setup_inputs() order:
    //  0 thresholds(f32,N) 1 rules(i32,N) 2 binary_ops(i32,N)
    //  3 left(i32,N) 4 right(i32,N) 5 leaf_is_reroll(bool,L)
    //  6..9 leaf_mask_* / leaf_score_cat (unused by reference outputs)
    const int* left  = (const int*)d_in[3];
    const int* right = (const int*)d_in[4];
    const unsigned char* leaf_is_reroll = (const unsigned char*)d_in[5];
    float* out = (float*)d_out;

    dag_genome_kernel<<<1, 256, 0, stream>>>(left, right, leaf_is_reroll, out);
}